// NoisyRNN_18940805776068
// MI455X (gfx1250) — compile-verified
//
#include <hip/hip_runtime.h>

typedef __attribute__((ext_vector_type(2))) float v2f;
typedef __attribute__((ext_vector_type(8))) float v8f;

#define N_UNITS     8
#define INPUT_DIM   64
#define OUT_CLASSES 2
#define BATCH       4096
#define T_STEPS     512
#define EPS_C       0.01f
#define BETA_C      0.8f
#define GAMMA_A_C   0.01f
#define GAMMA_W_C   0.01f
#define ALPHA_C     1.0f

#define ROWS_PER_WAVE   16          // WMMA M dimension
#define WAVES_PER_BLOCK 2
#define BLOCK_THREADS   (WAVES_PER_BLOCK * 32)

// CDNA5 hardware V_TANH_F32 transcendental (1 TRANS op vs ~30 VALU ops of
// libm tanh on the serial recurrence critical path).
__device__ __forceinline__ float fast_tanh(float v) {
#if __has_builtin(__builtin_amdgcn_tanhf)
    return __builtin_amdgcn_tanhf(v);
#elif __has_builtin(__builtin_amdgcn_tanh_f32)
    return __builtin_amdgcn_tanh_f32(v);
#else
    return tanhf(v);
#endif
}

// One wave32 handles 16 batch rows for all 512 timesteps.
// Encode (x[16x64] @ E_w^T[64x8]) is done with 16 chained
// V_WMMA_F32_16X16X4_F32 ops per step; the recurrence and decode are
// done with intra-wave permutes on the WMMA D-layout.
// NOTE: no waves-per-EU floor — a 256-VGPR cap makes the RA spill the
// software-pipelined loop into scratch (measured: vmem 78 -> 878), which is
// far worse in the hot serial loop than >256-VGPR MSB addressing (SALU-only).
__global__ __launch_bounds__(BLOCK_THREADS)
void noisy_rnn_fused(const float* __restrict__ x,
                     const float* __restrict__ E_w,
                     const float* __restrict__ E_b,
                     const float* __restrict__ D_w,
                     const float* __restrict__ D_b,
                     const float* __restrict__ B_mat,
                     const float* __restrict__ C_mat,
                     const float* __restrict__ d_vec,
                     float* __restrict__ out) {
    const int lane   = threadIdx.x & 31;
    const int waveId = blockIdx.x * WAVES_PER_BLOCK + (threadIdx.x >> 5);
    const int b0     = waveId * ROWS_PER_WAVE;

    const int  n     = lane & 15;   // WMMA column: unit index (valid if n < 8)
    const int  g     = lane >> 4;   // lane half: selects K pair (A) / row group (D)
    const bool valid = (n < N_UNITS);

    // ---- per-lane constants -------------------------------------------------
    // Columns of A and W for this lane's unit n:
    //   A = beta*(B - B^T) + (1-beta)*(B + B^T) - gamma_A*I   (W from C_mat)
    float Acol[N_UNITS], Wcol[N_UNITS];
#pragma unroll
    for (int m = 0; m < N_UNITS; ++m) {
        if (valid) {
            float bmn = B_mat[m * N_UNITS + n], bnm = B_mat[n * N_UNITS + m];
            float cmn = C_mat[m * N_UNITS + n], cnm = C_mat[n * N_UNITS + m];
            Acol[m] = BETA_C * (bmn - bnm) + (1.0f - BETA_C) * (bmn + bnm)
                      - ((m == n) ? GAMMA_A_C : 0.0f);
            Wcol[m] = BETA_C * (cmn - cnm) + (1.0f - BETA_C) * (cmn + cnm)
                      - ((m == n) ? GAMMA_W_C : 0.0f);
        } else {
            Acol[m] = 0.0f;
            Wcol[m] = 0.0f;
        }
    }
    const float ebn = valid ? E_b[n] : 0.0f;
    const float dn  = valid ? d_vec[n] : 0.0f;
    const float dw0 = valid ? D_w[0 * N_UNITS + n] : 0.0f;
    const float dw1 = valid ? D_w[1 * N_UNITS + n] : 0.0f;
    const float db0 = D_b[0], db1 = D_b[1];

    // B-operand fragments for the encode WMMA chain: B[k][n] = E_w[n][k].
    // 32-bit 4x16 B layout: lanes 0-15 hold K={0,1}, lanes 16-31 hold K={2,3}.
    // Invariant across timesteps -> hoist out of the t loop.
    v2f ew[INPUT_DIM / 4];
#pragma unroll
    for (int j = 0; j < INPUT_DIM / 4; ++j) {
        if (valid) {
            ew[j].x = E_w[n * INPUT_DIM + 4 * j + 2 * g];
            ew[j].y = E_w[n * INPUT_DIM + 4 * j + 2 * g + 1];
        } else {
            ew[j].x = 0.0f;
            ew[j].y = 0.0f;
        }
    }

    // A-operand addressing: this lane supplies row M = n (batch row b0+n),
    // K pair starting at 2*g within each K=4 chunk.
    const float* px = x + ((size_t)(b0 + n) * T_STEPS) * INPUT_DIM + 2 * g;

    // Hidden state: lane owns unit n for batch rows b0 + 8*g + r, r = 0..7
    // (matches the WMMA f32 C/D register layout exactly).
    float h[8];
#pragma unroll
    for (int r = 0; r < 8; ++r) h[r] = 0.0f;

#pragma unroll 1
    for (int t = 0; t < T_STEPS; ++t) {
        // ---- encode: z = x_tile @ E_w^T + E_b via chained WMMA --------------
        v8f acc;
#pragma unroll
        for (int i = 0; i < 8; ++i) acc[i] = ebn;

#pragma unroll
        for (int j = 0; j < INPUT_DIM / 4; ++j) {
            v2f a = *(const v2f*)(px + 4 * j);
            // (neg_a, A, neg_b, B, c_mod, C, reuse_a, reuse_b)
            acc = __builtin_amdgcn_wmma_f32_16x16x4_f32(
                false, a, false, ew[j], (short)0, acc, false, false);
        }
        px += INPUT_DIM;
        __builtin_prefetch(px, 0, 3);   // next timestep's row -> global_prefetch_b8

        // ---- recurrence + decode -------------------------------------------
#pragma unroll
        for (int r = 0; r < 8; ++r) {
            float hA = 0.0f, hW = 0.0f;
#pragma unroll
            for (int m = 0; m < N_UNITS; ++m) {
                float hm = __shfl(h[r], (lane & 16) + m, 32);  // h[b_r][m]
                hA = fmaf(hm, Acol[m], hA);
                hW = fmaf(hm, Wcol[m], hW);
            }
            float upd = ALPHA_C * hA + fast_tanh(hW + acc[r]);
            float hn  = h[r] + dn * upd;
            h[r] = hn;

            // decode: out[b_r][t][c] = sum_n h*D_w[c][n] + D_b[c]
            float s0 = hn * dw0;
            float s1 = hn * dw1;
#pragma unroll
            for (int off = 1; off < N_UNITS; off <<= 1) {
                s0 += __shfl_xor(s0, off, 32);
                s1 += __shfl_xor(s1, off, 32);
            }
            if (n == 0) {
                size_t b = (size_t)(b0 + 8 * g + r);
                float2 o = make_float2(s0 + db0, s1 + db1);
                *(float2*)(out + (b * T_STEPS + t) * OUT_CLASSES) = o;
            }
        }
    }
}

extern "C" void kernel_launch(void* const* d_in, const int* in_sizes, int n_in,
                              void* d_out, int out_size, void* d_ws, size_t ws_size,
                              hipStream_t stream) {
    (void)in_sizes; (void)n_in; (void)d_ws; (void)ws_size; (void)out_size;
    const float* x     = (const float*)d_in[0];
    const float* E_w   = (const float*)d_in[1];
    const float* E_b   = (const float*)d_in[2];
    const float* D_w   = (const float*)d_in[3];
    const float* D_b   = (const float*)d_in[4];
    const float* B_mat = (const float*)d_in[5];
    const float* C_mat = (const float*)d_in[6];
    const float* d_vec = (const float*)d_in[7];
    float* out = (float*)d_out;

    dim3 grid(BATCH / (ROWS_PER_WAVE * WAVES_PER_BLOCK));   // 128 blocks
    dim3 block(BLOCK_THREADS);                              // 64 threads = 2 waves
    noisy_rnn_fused<<<grid, block, 0, stream>>>(
        x, E_w, E_b, D_w, D_b, B_mat, C_mat, d_vec, out);
}